// DocumentGNN_43207370998208
// MI455X (gfx1250) — compile-verified
//
#include <hip/hip_runtime.h>
#include <hip/hip_bf16.h>
#include <math.h>
#include <stdint.h>

#define NUM_GRAPHS 1024

typedef float v2f __attribute__((ext_vector_type(2)));
typedef float v8f __attribute__((ext_vector_type(8)));

// ---------------- degree / normalization ----------------
__global__ void k_deg_init(float* deg, int N) {
  int i = blockIdx.x * blockDim.x + threadIdx.x;
  if (i < N) deg[i] = 1.0f;  // self-loop contributes 1
}

__global__ void k_deg_count(const int* __restrict__ col, float* deg, int E) {
  int e = blockIdx.x * blockDim.x + threadIdx.x;
  if (e < E) {
    int c = __builtin_nontemporal_load(&col[e]);  // single-use stream: don't pollute L2
    atomicAdd(&deg[c], 1.0f);
  }
}

__global__ void k_dinv(const float* __restrict__ deg, float* __restrict__ dinv, int N) {
  int i = blockIdx.x * blockDim.x + threadIdx.x;
  if (i < N) dinv[i] = rsqrtf(deg[i]);
}

// ---------------- layer-1 transform: xw1[N,16] = x[N,3] @ W1[3,16] ----------------
// One wave per 16-node tile, single V_WMMA_F32_16X16X4_F32 (K padded 3 -> 4).
__global__ __launch_bounds__(256) void k_xw1(const float* __restrict__ x,
                                             const float* __restrict__ W1,
                                             float* __restrict__ xw1,
                                             int ntiles, int N) {
  int wave = (blockIdx.x * blockDim.x + threadIdx.x) >> 5;
  int lane = threadIdx.x & 31;
  if (wave >= ntiles) return;  // uniform per-wave: EXEC stays all-ones for WMMA

  int hi = lane >> 4;          // 0: lanes 0-15, 1: lanes 16-31
  int k0 = hi * 2;             // A holds K = {0,1} or {2,3}
  int n  = lane & 15;
  int m  = wave * 16 + n;      // A: M = lane%16
  int mc = (m < N) ? m : (N - 1);  // clamp: pad-lane loads are safe, outputs unguarded-stored only on full tiles

  v2f a;
  a.x = x[(size_t)mc * 3 + k0];                             // K=0 or K=2
  a.y = (k0 + 1 < 3) ? x[(size_t)mc * 3 + k0 + 1] : 0.f;    // K=1; K=3 is pad
  v2f b;
  b.x = W1[k0 * 16 + n];
  b.y = (k0 + 1 < 3) ? W1[(k0 + 1) * 16 + n] : 0.f;

  v8f c = {};
  c = __builtin_amdgcn_wmma_f32_16x16x4_f32(false, a, false, b, (short)0, c, false, false);

  // D: VGPR r -> row M = r + hi*8, col = n. One base + immediate offsets.
  int m0 = wave * 16 + hi * 8;
  float* base = xw1 + (size_t)m0 * 16 + n;
  if (wave * 16 + 16 <= N) {           // wave-uniform: full tile, no guards
#pragma unroll
    for (int r = 0; r < 8; ++r) base[(size_t)r * 16] = c[r];
  } else {                             // tail tile only
#pragma unroll
    for (int r = 0; r < 8; ++r)
      if (m0 + r < N) base[(size_t)r * 16] = c[r];
  }
}

// ---------------- layer-2 transform: xw2[N,32] = relu(out1+b1)[N,16] @ W2[16,32] ----
// Bias+ReLU of layer 1 fused into the A-matrix load. K=16 = chain of 4 f32 WMMAs,
// two N-tiles of 16 -> 8 WMMAs per wave-tile.
__global__ __launch_bounds__(256) void k_xw2(const float* __restrict__ out1,
                                             const float* __restrict__ b1,
                                             const float* __restrict__ W2,
                                             float* __restrict__ xw2,
                                             int ntiles, int N) {
  int wave = (blockIdx.x * blockDim.x + threadIdx.x) >> 5;
  int lane = threadIdx.x & 31;
  if (wave >= ntiles) return;

  int hi = lane >> 4;
  int n  = lane & 15;
  int m  = wave * 16 + n;
  int mc = (m < N) ? m : (N - 1);
  const float* arow = out1 + (size_t)mc * 16;

  v2f a[4];
#pragma unroll
  for (int kc = 0; kc < 4; ++kc) {
    int k0 = kc * 4 + hi * 2;
    a[kc].x = fmaxf(arow[k0] + b1[k0], 0.0f);          // fused bias+ReLU (h1)
    a[kc].y = fmaxf(arow[k0 + 1] + b1[k0 + 1], 0.0f);
  }

  int m0 = wave * 16 + hi * 8;
  bool full = (wave * 16 + 16 <= N);    // wave-uniform
#pragma unroll
  for (int nb = 0; nb < 2; ++nb) {
    v8f c = {};
#pragma unroll
    for (int kc = 0; kc < 4; ++kc) {
      int k0 = kc * 4 + hi * 2;
      v2f b;
      b.x = W2[(size_t)k0 * 32 + nb * 16 + n];
      b.y = W2[(size_t)(k0 + 1) * 32 + nb * 16 + n];
      c = __builtin_amdgcn_wmma_f32_16x16x4_f32(false, a[kc], false, b, (short)0, c, false, false);
    }
    float* base = xw2 + (size_t)m0 * 32 + nb * 16 + n;
    if (full) {
#pragma unroll
      for (int r = 0; r < 8; ++r) base[(size_t)r * 32] = c[r];
    } else {
#pragma unroll
      for (int r = 0; r < 8; ++r)
        if (m0 + r < N) base[(size_t)r * 32] = c[r];
    }
  }
}

// ---------------- self-loop init: out[i,:] = dinv[i]^2 * xw[i,:] (float4) --------
template <int F>
__global__ void k_selfloop(const float4* __restrict__ xw, const float* __restrict__ dinv,
                           float4* __restrict__ out, int N) {
  int idx = blockIdx.x * blockDim.x + threadIdx.x;   // over N * F/4 float4s
  if (idx < N * (F / 4)) {
    int i = idx / (F / 4);
    float d = dinv[i];
    float s = d * d;
    float4 v = xw[idx];
    v.x *= s; v.y *= s; v.z *= s; v.w *= s;
    out[idx] = v;
  }
}

// ---------------- edge scatter: out[col,:] += dinv[row]*dinv[col] * xw[row,:] ----
// One thread per edge. Edge indices streamed non-temporally (single use); gathers of
// xw rows hit the 192MB L2 (tables are 12.8/25.6MB); scatters are f32 L2 atomics.
template <int F>
__global__ void k_edge_agg(const int* __restrict__ row, const int* __restrict__ col,
                           const float* __restrict__ dinv, const float* __restrict__ xw,
                           float* out, int E) {
  int e = blockIdx.x * blockDim.x + threadIdx.x;
  if (e >= E) return;
  int r = __builtin_nontemporal_load(&row[e]);
  int c = __builtin_nontemporal_load(&col[e]);
  float norm = dinv[r] * dinv[c];
  const float4* src = (const float4*)(xw + (size_t)r * F);
  float4 m[F / 4];
#pragma unroll
  for (int v = 0; v < F / 4; ++v) m[v] = src[v];     // issue all gathers first
  float* dst = out + (size_t)c * F;
#pragma unroll
  for (int v = 0; v < F / 4; ++v) {
    atomicAdd(dst + 4 * v + 0, m[v].x * norm);
    atomicAdd(dst + 4 * v + 1, m[v].y * norm);
    atomicAdd(dst + 4 * v + 2, m[v].z * norm);
    atomicAdd(dst + 4 * v + 3, m[v].w * norm);
  }
}

// ---------------- pooling ----------------
__global__ void k_zero_pool(float4* gsum4, float* gcnt) {
  int i = blockIdx.x * blockDim.x + threadIdx.x;
  if (i < NUM_GRAPHS * 8) gsum4[i] = make_float4(0.f, 0.f, 0.f, 0.f);
  if (i < NUM_GRAPHS) gcnt[i] = 0.0f;
}

__global__ void k_pool_feat(const float* __restrict__ o2, const float* __restrict__ b2,
                            const int* __restrict__ batch, float* gsum, int N) {
  int idx = blockIdx.x * blockDim.x + threadIdx.x;
  if (idx >= N * 32) return;
  int i = idx >> 5, f = idx & 31;
  float v = fmaxf(o2[idx] + b2[f], 0.0f);  // h2 = relu(gcn2) fused, never materialized
  atomicAdd(&gsum[batch[i] * 32 + f], v);
}

__global__ void k_pool_cnt(const int* __restrict__ batch, float* gcnt, int N) {
  int i = blockIdx.x * blockDim.x + threadIdx.x;
  if (i < N) atomicAdd(&gcnt[batch[i]], 1.0f);
}

// ---------------- head: logits = mean @ Wfc + bfc ; log_softmax ----------------
__global__ void k_head(const float* __restrict__ gsum, const float* __restrict__ gcnt,
                       const float* __restrict__ Wfc, const float* __restrict__ bfc,
                       float* __restrict__ out) {
  int g = blockIdx.x * blockDim.x + threadIdx.x;
  if (g >= NUM_GRAPHS) return;
  float inv = 1.0f / fmaxf(gcnt[g], 1.0f);
  float l0 = bfc[0], l1 = bfc[1];
#pragma unroll
  for (int k = 0; k < 32; ++k) {
    float m = gsum[g * 32 + k] * inv;
    l0 += m * Wfc[k * 2 + 0];
    l1 += m * Wfc[k * 2 + 1];
  }
  float mx = fmaxf(l0, l1);
  float lse = mx + logf(expf(l0 - mx) + expf(l1 - mx));
  out[g * 2 + 0] = l0 - lse;
  out[g * 2 + 1] = l1 - lse;
}

extern "C" void kernel_launch(void* const* d_in, const int* in_sizes, int n_in,
                              void* d_out, int out_size, void* d_ws, size_t ws_size,
                              hipStream_t stream) {
  const float* x     = (const float*)d_in[0];
  const int*   ei    = (const int*)d_in[1];   // [2, E]: row (sources), col (targets)
  const int*   batch = (const int*)d_in[2];
  const float* W1    = (const float*)d_in[3];
  const float* b1    = (const float*)d_in[4];
  const float* W2    = (const float*)d_in[5];
  const float* b2    = (const float*)d_in[6];
  const float* Wfc   = (const float*)d_in[7];
  const float* bfc   = (const float*)d_in[8];

  const int N = in_sizes[0] / 3;
  const int E = in_sizes[1] / 2;
  const int* row = ei;
  const int* col = ei + E;

  // Carve workspace (256B-aligned regions).
  char* p = (char*)d_ws;
  auto carve = [&](size_t nbytes) -> float* {
    uintptr_t q = ((uintptr_t)p + 255) & ~(uintptr_t)255;
    p = (char*)(q + nbytes);
    return (float*)q;
  };
  float* deg  = carve((size_t)N * 4);
  float* dinv = carve((size_t)N * 4);
  float* xw1  = carve((size_t)N * 16 * 4);
  float* o1   = carve((size_t)N * 16 * 4);   // aggregated layer-1 pre-activation
  float* xw2  = carve((size_t)N * 32 * 4);
  float* o2   = carve((size_t)N * 32 * 4);   // aggregated layer-2 pre-activation
  float* gsum = carve((size_t)NUM_GRAPHS * 32 * 4);
  float* gcnt = carve((size_t)NUM_GRAPHS * 4);
  (void)ws_size; (void)n_in; (void)out_size;

  const int B = 256;
  auto cdiv = [](long long a, long long b) { return (int)((a + b - 1) / b); };
  const int ntiles = cdiv(N, 16);

  // Degrees + symmetric normalization
  k_deg_init <<<cdiv(N, B), B, 0, stream>>>(deg, N);
  k_deg_count<<<cdiv(E, B), B, 0, stream>>>(col, deg, E);
  k_dinv     <<<cdiv(N, B), B, 0, stream>>>(deg, dinv, N);

  // Layer 1: transform (WMMA), self-loop init, edge scatter
  k_xw1          <<<cdiv(ntiles, 8), B, 0, stream>>>(x, W1, xw1, ntiles, N);
  k_selfloop<16> <<<cdiv((long long)N * 4, B), B, 0, stream>>>((const float4*)xw1, dinv,
                                                               (float4*)o1, N);
  k_edge_agg<16> <<<cdiv(E, B), B, 0, stream>>>(row, col, dinv, xw1, o1, E);

  // Layer 2: transform (WMMA chain, bias+relu of layer 1 fused into A load)
  k_xw2          <<<cdiv(ntiles, 8), B, 0, stream>>>(o1, b1, W2, xw2, ntiles, N);
  k_selfloop<32> <<<cdiv((long long)N * 8, B), B, 0, stream>>>((const float4*)xw2, dinv,
                                                               (float4*)o2, N);
  k_edge_agg<32> <<<cdiv(E, B), B, 0, stream>>>(row, col, dinv, xw2, o2, E);

  // Pool (bias+relu fused) + head
  k_zero_pool<<<cdiv(NUM_GRAPHS * 8, B), B, 0, stream>>>((float4*)gsum, gcnt);
  k_pool_feat<<<cdiv((long long)N * 32, B), B, 0, stream>>>(o2, b2, batch, gsum, N);
  k_pool_cnt <<<cdiv(N, B), B, 0, stream>>>(batch, gcnt, N);
  k_head     <<<cdiv(NUM_GRAPHS, B), B, 0, stream>>>(gsum, gcnt, Wfc, bfc, (float*)d_out);
}